// EfficientMemoryGELU_45981919871058
// MI455X (gfx1250) — compile-verified
//
#include <hip/hip_runtime.h>
#include <hip/hip_bf16.h>

// Exact-erf GELU, fp32 -> fp32, pure streaming elementwise op.
//
// Roofline on MI455X: 512 MiB of traffic @ 23.3 TB/s ~= 23 us; ~1.2 GFLOP of
// VALU work is negligible next to that. So the whole optimization problem is
// "move 128-bit lines with non-temporal hints at full occupancy". WMMA is
// intentionally NOT used: there is no matmul structure to map onto v_wmma_*.

typedef float float4v __attribute__((ext_vector_type(4)));

__device__ __forceinline__ float gelu_exact(float x) {
    // 0.5 * x * (1 + erf(x / sqrt(2)))
    const float kInvSqrt2 = 0.70710678118654752440f;
    return 0.5f * x * (1.0f + erff(x * kInvSqrt2));
}

__global__ void __launch_bounds__(256)
gelu_exact_f32_vec4_kernel(const float* __restrict__ x,
                           float* __restrict__ y,
                           long long n) {
    const long long nvec   = n >> 2;                       // # of float4 packets
    const long long stride = (long long)gridDim.x * blockDim.x;
    long long i = (long long)blockIdx.x * blockDim.x + threadIdx.x;

    const float4v* __restrict__ xv = (const float4v*)x;
    float4v* __restrict__ yv       = (float4v*)y;

    for (; i < nvec; i += stride) {
        // Prefetch the next grid-stride iteration's line (gfx1250
        // global_prefetch_b8; speculative, no counter cost).
        if (i + stride < nvec) {
            __builtin_prefetch(&xv[i + stride], 0, 0);
        }
        // Non-temporal 128-bit load: streaming data, never re-read; keep it
        // out of L2/WGP$ residency (TH=NT).
        float4v v = __builtin_nontemporal_load(&xv[i]);
        float4v r;
#pragma unroll
        for (int k = 0; k < 4; ++k) {
            r[k] = gelu_exact(v[k]);
        }
        // Non-temporal 128-bit store: write-streaming, no re-read.
        __builtin_nontemporal_store(r, &yv[i]);
    }

    // Scalar tail for n % 4 != 0 (not hit for 4*4096*4096, kept for
    // correctness on arbitrary sizes). Handled by the first few threads of
    // the grid only.
    const long long tail_start = nvec << 2;
    long long t = tail_start + (long long)blockIdx.x * blockDim.x + threadIdx.x;
    if (blockIdx.x == 0 && t < n) {
        // blockIdx.x==0 guard keeps this single-pass; tail < 4 elements so
        // threads 0..tail-1 of block 0 cover it.
        long long idx = tail_start + threadIdx.x;
        if (idx < n) {
            y[idx] = gelu_exact(x[idx]);
        }
    }
}

extern "C" void kernel_launch(void* const* d_in, const int* in_sizes, int n_in,
                              void* d_out, int out_size, void* d_ws, size_t ws_size,
                              hipStream_t stream) {
    (void)n_in; (void)d_ws; (void)ws_size;

    const float* x = (const float*)d_in[0];
    float* y       = (float*)d_out;
    const long long n = (long long)in_sizes[0];   // 4*4096*4096 = 67,108,864

    const int threads = 256;                       // 8 wave32 waves per block
    const long long nvec = n >> 2;
    // ~4 float4 iterations per thread: enough blocks to fill every SIMD32
    // many times over, few enough that the per-thread loop lets the compiler
    // software-pipeline loads ahead of the erf math.
    long long blocks = (nvec + (long long)threads * 4 - 1) / ((long long)threads * 4);
    if (blocks < 1) blocks = 1;
    if (blocks > 1048576) blocks = 1048576;

    gelu_exact_f32_vec4_kernel<<<(uint32_t)blocks, threads, 0, stream>>>(x, y, n);
}